// BoundaryLoss_49306224558104
// MI455X (gfx1250) — compile-verified
//
#include <hip/hip_runtime.h>
#include <hip/hip_bf16.h>

// BoundaryLoss for MI455X (gfx1250, wave32).
//
// Key algebraic reduction: in the reference, edt(fg)==0 on all bg pixels and
// edt(!fg)==0 on all fg pixels, so min(pos,neg) == 0 everywhere and
// dist_map == 0 identically. The loss collapses to
//     mean( softplus(x) - x*z ),  x = pred logits, z = (target>0).
// That is a 4 MB-read, 1-float-write reduction: HBM roofline 4MB/23.3TB/s
// ~ 0.17us. We therefore optimize purely for clean B128 streaming loads and
// a deterministic two-pass fp32 reduction.
//
// CDNA5 path: the wave32 cross-lane sum is done on the matrix pipe with
// V_WMMA_F32_16X16X4_F32 (ones-matrix reduction, fp32 throughout):
//   A[16x4]: lane L holds its partial in K=0 (L<16) / K=2 (L>=16), rest 0
//   B[4x16]: all ones  ->  D[m][n] = s_m + s_{m+16} for every n
// Summing the 8 C/D VGPRs per lane + one shfl_xor(16) yields the wave total.
// EXEC is all-ones at both call sites (required by WMMA).

typedef __attribute__((ext_vector_type(2))) float v2f;
typedef __attribute__((ext_vector_type(8))) float v8f;

#define N_ELEMS (8 * 256 * 256)   // 524288
#define BLOCKS  256
#define THREADS 256
#define WAVES   (THREADS / 32)

// Full-wave (32-lane) fp32 sum via the matrix unit. Returns total in all lanes.
__device__ __forceinline__ float wave_sum_wmma(float s) {
    v2f a; a[0] = s;    a[1] = 0.0f;   // A 16x4: K=0 (lanes 0-15) / K=2 (lanes 16-31)
    v2f b; b[0] = 1.0f; b[1] = 1.0f;   // B 4x16: all ones (layout-independent)
    v8f c = {};
    // (neg_a, A, neg_b, B, c_mod, C, reuse_a, reuse_b)
    c = __builtin_amdgcn_wmma_f32_16x16x4_f32(false, a, false, b, (short)0, c,
                                              false, false);
    // D[m][n] identical across n; per-lane sum over the 8 row-VGPRs covers
    // m=0..7 (lanes 0-15) or m=8..15 (lanes 16-31).
    float t = c[0] + c[1] + c[2] + c[3] + c[4] + c[5] + c[6] + c[7];
    t += __shfl_xor(t, 16, 32);        // combine the two m-halves
    return t;
}

// Numerically stable BCE-with-logits, matching jax.nn.softplus(x) - x*z:
// softplus(x) = max(x,0) + log1p(exp(-|x|)).
__device__ __forceinline__ float bce_logits(float x, float z) {
    return fmaxf(x, 0.0f) + log1pf(expf(-fabsf(x))) - x * z;
}

__global__ void __launch_bounds__(THREADS)
bce_partial_kernel(const float* __restrict__ pred,
                   const int*   __restrict__ target,
                   float*       __restrict__ partial) {
    __shared__ float smem[WAVES];

    const int tid      = blockIdx.x * THREADS + threadIdx.x;
    const int nthreads = gridDim.x * THREADS;
    const int n4       = N_ELEMS / 4;

    const float4* __restrict__ p4 = (const float4*)pred;
    const int4*   __restrict__ t4 = (const int4*)target;

    float s = 0.0f;
    for (int i = tid; i < n4; i += nthreads) {     // exactly 2 iters/thread
        float4 x = p4[i];
        int4   t = t4[i];
        s += bce_logits(x.x, t.x > 0 ? 1.0f : 0.0f);
        s += bce_logits(x.y, t.y > 0 ? 1.0f : 0.0f);
        s += bce_logits(x.z, t.z > 0 ? 1.0f : 0.0f);
        s += bce_logits(x.w, t.w > 0 ? 1.0f : 0.0f);
    }

    // Matrix-pipe wave reduction (uniform control flow: EXEC all ones here).
    float ws = wave_sum_wmma(s);

    const int wave = threadIdx.x >> 5;
    if ((threadIdx.x & 31) == 0) smem[wave] = ws;
    __syncthreads();
    if (threadIdx.x == 0) {
        float bsum = 0.0f;
        #pragma unroll
        for (int w = 0; w < WAVES; ++w) bsum += smem[w];
        partial[blockIdx.x] = bsum;
    }
}

__global__ void __launch_bounds__(THREADS)
bce_finalize_kernel(const float* __restrict__ partial,
                    float*       __restrict__ out) {
    __shared__ float smem[WAVES];

    float s = partial[threadIdx.x];                // THREADS == BLOCKS == 256
    float ws = wave_sum_wmma(s);                   // EXEC all ones

    const int wave = threadIdx.x >> 5;
    if ((threadIdx.x & 31) == 0) smem[wave] = ws;
    __syncthreads();
    if (threadIdx.x == 0) {
        float total = 0.0f;
        #pragma unroll
        for (int w = 0; w < WAVES; ++w) total += smem[w];
        out[0] = total * (1.0f / (float)N_ELEMS); // mean; dist_map term == 0
    }
}

extern "C" void kernel_launch(void* const* d_in, const int* in_sizes, int n_in,
                              void* d_out, int out_size, void* d_ws, size_t ws_size,
                              hipStream_t stream) {
    const float* pred   = (const float*)d_in[0]; // [8,1,256,256] f32
    const int*   target = (const int*)d_in[1];   // [8,256,256]   i32
    float*       out    = (float*)d_out;         // scalar f32
    float*       part   = (float*)d_ws;          // BLOCKS f32 partials (1 KB)

    bce_partial_kernel <<<BLOCKS, THREADS, 0, stream>>>(pred, target, part);
    bce_finalize_kernel<<<1,      THREADS, 0, stream>>>(part, out);
}